// GDCCostVolumeChunkedBCDHW_67697274520018
// MI455X (gfx1250) — compile-verified
//
#include <hip/hip_runtime.h>
#include <math.h>

// ---- problem shape (fixed by setup_inputs) ----
#define BATCH 2
#define CIN   64
#define HH    128
#define WW    256
#define HW    (HH * WW)      // 32768
#define OCH   96             // COST_CH
#define GRP   8
#define CG    12             // OCH / GRP
#define DP    32             // ceil(128/4)
#define TAUC  0.07f

typedef __attribute__((ext_vector_type(2))) float v2f;
typedef __attribute__((ext_vector_type(8))) float v8f;

// =====================================================================
// Kernel 1: per-pixel projection  P[96] = Wm(96x64) @ F[64]  then L2-norm
// over the 96 output channels.  One wave32 handles one block of 16 pixels.
// GEMM done with V_WMMA_F32_16X16X4_F32: 6 M-tiles (96 rows) x 16 K-steps.
//
// Operand layout (ISA 7.12.2, 32-bit):
//   A 16x4 : lane l -> M = l&15,  K = 4k + 2*(l>>4) + {0,1} (2 VGPRs)
//   B 4x16 : lane l -> N = l&15,  K = 4k + 2*(l>>4) + {0,1} (2 VGPRs)
//   C/D    : lane l -> N = l&15,  M = 8*(l>>4) + vgpr_idx   (8 VGPRs)
// =====================================================================
__global__ __launch_bounds__(256) void proj_norm(const float* __restrict__ F,
                                                 const float* __restrict__ Wm,
                                                 float* __restrict__ P)
{
    const int lane = threadIdx.x & 31;
    const int wave = threadIdx.x >> 5;
    const int pb   = blockIdx.x * 8 + wave;        // 16-pixel block id, 0..4095
    const int pabs = pb << 4;
    const int b    = pabs / HW;
    const int p0   = pabs - b * HW;
    const int n    = lane & 15;                    // N / pixel column, M row
    const int koff = (lane >> 4) << 1;             // 0 or 2 (K half-split)

    const float* __restrict__ Fb = F + (size_t)b * CIN * HW + p0 + n;

    __builtin_prefetch(Fb, 0, 0);                  // global_prefetch_b8

    v8f acc0 = {}, acc1 = {}, acc2 = {}, acc3 = {}, acc4 = {}, acc5 = {};

    #pragma unroll
    for (int k = 0; k < 16; ++k) {
        const int c = 4 * k + koff;
        v2f bfrag;
        bfrag.x = Fb[(size_t)c * HW];
        bfrag.y = Fb[(size_t)(c + 1) * HW];

        const float* wr = Wm + n * CIN + c;        // tile t row: + t*16*CIN
        v2f a0 = *(const v2f*)(wr + 0 * 16 * CIN);
        v2f a1 = *(const v2f*)(wr + 1 * 16 * CIN);
        v2f a2 = *(const v2f*)(wr + 2 * 16 * CIN);
        v2f a3 = *(const v2f*)(wr + 3 * 16 * CIN);
        v2f a4 = *(const v2f*)(wr + 4 * 16 * CIN);
        v2f a5 = *(const v2f*)(wr + 5 * 16 * CIN);

        acc0 = __builtin_amdgcn_wmma_f32_16x16x4_f32(false, a0, false, bfrag, (short)0, acc0, false, false);
        acc1 = __builtin_amdgcn_wmma_f32_16x16x4_f32(false, a1, false, bfrag, (short)0, acc1, false, false);
        acc2 = __builtin_amdgcn_wmma_f32_16x16x4_f32(false, a2, false, bfrag, (short)0, acc2, false, false);
        acc3 = __builtin_amdgcn_wmma_f32_16x16x4_f32(false, a3, false, bfrag, (short)0, acc3, false, false);
        acc4 = __builtin_amdgcn_wmma_f32_16x16x4_f32(false, a4, false, bfrag, (short)0, acc4, false, false);
        acc5 = __builtin_amdgcn_wmma_f32_16x16x4_f32(false, a5, false, bfrag, (short)0, acc5, false, false);
    }

    // L2 norm over 96 channels of pixel n: this lane holds 48 of them,
    // lane^16 holds the other 48.
    float ss = 0.f;
    #pragma unroll
    for (int r = 0; r < 8; ++r) {
        ss += acc0[r] * acc0[r];
        ss += acc1[r] * acc1[r];
        ss += acc2[r] * acc2[r];
        ss += acc3[r] * acc3[r];
        ss += acc4[r] * acc4[r];
        ss += acc5[r] * acc5[r];
    }
    ss += __shfl_xor(ss, 16, 32);
    const float scale = 1.0f / fmaxf(sqrtf(ss), 1e-12f);

    float* __restrict__ Pb = P + (size_t)b * OCH * HW + p0 + n;
    const int mh = (lane >> 4) << 3;               // 0 or 8
    #pragma unroll
    for (int r = 0; r < 8; ++r) {
        Pb[(size_t)( 0 + mh + r) * HW] = acc0[r] * scale;
        Pb[(size_t)(16 + mh + r) * HW] = acc1[r] * scale;
        Pb[(size_t)(32 + mh + r) * HW] = acc2[r] * scale;
        Pb[(size_t)(48 + mh + r) * HW] = acc3[r] * scale;
        Pb[(size_t)(64 + mh + r) * HW] = acc4[r] * scale;
        Pb[(size_t)(80 + mh + r) * HW] = acc5[r] * scale;
    }
}

// =====================================================================
// Kernel 2: grouped shifted correlation + sin bias + scaling.
// Block = (b, g, h, 64-wide w chunk); all 32 disparities.
// LDS stages FL (12x64) and the left-clamped FR window (12x96).
// cost[b,g,d,h,w] = dot12 * (-1/(sqrt(12)*tau)) + sin(2*pi*d/32) * (-1/tau)
// =====================================================================
__global__ __launch_bounds__(256) void cost_volume(const float* __restrict__ PL,
                                                   const float* __restrict__ PR,
                                                   float* __restrict__ out)
{
    __shared__ float sFL[CG][64];
    __shared__ float sFR[CG][97];   // +1 pad

    const int bx = blockIdx.x;      // h*4 + wchunk
    const int wc = bx & 3;
    const int h  = bx >> 2;
    const int g  = blockIdx.y;
    const int b  = blockIdx.z;
    const int w0 = wc * 64;
    const int tid = threadIdx.x;

    const float* __restrict__ baseL = PL + ((size_t)b * OCH + g * CG) * HW + h * WW;
    const float* __restrict__ baseR = PR + ((size_t)b * OCH + g * CG) * HW + h * WW;

    for (int i = tid; i < CG * 64; i += 256) {
        const int c = i >> 6, x = i & 63;
        sFL[c][x] = baseL[(size_t)c * HW + w0 + x];
    }
    for (int i = tid; i < CG * 96; i += 256) {
        const int c = i / 96, j = i - c * 96;
        int gw = w0 - 32 + j;               // global w for shifted read
        gw = gw < 0 ? 0 : gw;               // clip(w-d, 0, W-1): upper never hit
        sFR[c][j] = baseR[(size_t)c * HW + gw];
    }
    __syncthreads();

    const int wl = tid & 63;
    const int d0 = tid >> 6;                // 0..3; thread covers d0 + 4*dd
    const float simScale = -0.28867513459481288f / TAUC;   // -(1/sqrt(12))/tau
    const float psiScale = -1.0f / TAUC;

    float* __restrict__ ob = out + (((size_t)b * GRP + g) * DP * HH + h) * WW + w0 + wl;

    #pragma unroll
    for (int dd = 0; dd < 8; ++dd) {
        const int d = d0 + (dd << 2);
        float s = 0.f;
        #pragma unroll
        for (int c = 0; c < CG; ++c)
            s += sFL[c][wl] * sFR[c][wl + 32 - d];
        const float psi = __sinf(6.28318530717958647692f * (float)d * (1.0f / 32.0f));
        ob[(size_t)d * HW] = s * simScale + psi * psiScale;
    }
}

// =====================================================================
extern "C" void kernel_launch(void* const* d_in, const int* in_sizes, int n_in,
                              void* d_out, int out_size, void* d_ws, size_t ws_size,
                              hipStream_t stream)
{
    const float* F_L = (const float*)d_in[0];
    const float* F_R = (const float*)d_in[1];
    const float* WL  = (const float*)d_in[2];
    const float* WR  = (const float*)d_in[3];
    float* out = (float*)d_out;

    float* PL = (float*)d_ws;                              // [B,96,HW]
    float* PR = PL + (size_t)BATCH * OCH * HW;             // [B,96,HW]

    // 4096 pixel-blocks per side, 8 waves per 256-thread block
    proj_norm<<<512, 256, 0, stream>>>(F_L, WL, PL);
    proj_norm<<<512, 256, 0, stream>>>(F_R, WR, PR);

    dim3 grid(HH * 4, GRP, BATCH);
    cost_volume<<<grid, 256, 0, stream>>>(PL, PR, out);
}